// QuantizationLayer_89507118449249
// MI455X (gfx1250) — compile-verified
//
#include <hip/hip_runtime.h>
#include <hip/hip_bf16.h>
#include <math.h>

typedef float v2f __attribute__((ext_vector_type(2)));
typedef float v8f __attribute__((ext_vector_type(8)));

#define NUM_CLASSES   100
#define KC            64
#define DIM           128
#define BATCH         32768
#define K_CENTERS     6400          // 100 * 64
#define CHUNK         64            // centers per LDS buffer
#define NCHUNK        (K_CENTERS / CHUNK)   // 100
#define CPAD          130           // padded LDS row stride (floats) to spread banks
#define WAVES         8
#define ROWS_PER_WAVE 16
#define BLOCK_ROWS    (WAVES * ROWS_PER_WAVE)   // 128 rows per block

// --- CDNA5 async global->LDS path (types per hipcc diagnostic: v4i pointers) ---
typedef int v4i_vs __attribute__((__vector_size__(16)));
typedef __attribute__((address_space(1))) v4i_vs* gv4i_ptr;   // global int4*
typedef __attribute__((address_space(3))) v4i_vs* lv4i_ptr;   // LDS int4*

// ---------------------------------------------------------------------------
// Pass 1: ||center||^2 for all 6400 centers -> workspace
// ---------------------------------------------------------------------------
__global__ __launch_bounds__(256) void c2_kernel(const float* __restrict__ centers,
                                                 float* __restrict__ c2) {
    int k = blockIdx.x * blockDim.x + threadIdx.x;
    if (k >= K_CENTERS) return;
    const float4* p = (const float4*)(centers + (size_t)k * DIM);
    float s = 0.f;
#pragma unroll
    for (int i = 0; i < DIM / 4; ++i) {
        float4 v = p[i];
        s += v.x * v.x + v.y * v.y + v.z * v.z + v.w * v.w;
    }
    c2[k] = s;
}

// ---------------------------------------------------------------------------
// Pass 2: fused distance-argmin + gather, fp32 WMMA core, async double buffer
// ---------------------------------------------------------------------------
__global__ __launch_bounds__(256) void vq_kernel(const float* __restrict__ x,
                                                 const float* __restrict__ centers,
                                                 const float* __restrict__ c2,
                                                 float* __restrict__ outQ,
                                                 float* __restrict__ outClu,
                                                 float* __restrict__ outMin,
                                                 float* __restrict__ outCls) {
    __shared__ float sC[2][CHUNK * CPAD];   // 2 x 33.3 KB center chunk (padded)
    __shared__ float sC2[2][CHUNK];         // chunk ||c||^2
    __shared__ float sFinV[WAVES * 16];     // final per-row best value
    __shared__ int   sFinI[WAVES * 16];     // final per-row best index

    const int  tid    = threadIdx.x;
    const int  wave   = tid >> 5;
    const int  lane   = tid & 31;
    const int  laneLo = lane & 15;
    const bool hi     = lane >= 16;

    const int rowBase = blockIdx.x * BLOCK_ROWS + wave * ROWS_PER_WAVE;
    const int myRow   = rowBase + laneLo;

    // A panel in registers, pre-arranged in WMMA 16x4 f32 fragment order:
    // lanes 0-15 hold dims (4j, 4j+1) of row laneLo; lanes 16-31 hold (4j+2, 4j+3).
    v2f ax[32];
    const float* xr  = x + (size_t)myRow * DIM;
    float        x2p = 0.f;
#pragma unroll
    for (int j = 0; j < 32; ++j) {
        v2f t = *(const v2f*)(xr + 4 * j + (hi ? 2 : 0));
        ax[j] = t;
        x2p += t[0] * t[0] + t[1] * t[1];
    }
    float x2 = x2p + __shfl_xor(x2p, 16, 32);  // full ||x_row||^2 of row laneLo

    float bestV[8];
    int   bestI[8];
#pragma unroll
    for (int j = 0; j < 8; ++j) { bestV[j] = 3.4e38f; bestI[j] = 0; }

    // --- staging helper: 64 centers x 128 floats = 2048 int4, 8 async ops/thread ---
    auto stage = [&](int c0, int buf) {
        const float4* src = (const float4*)(centers + (size_t)c0 * DIM);
#pragma unroll
        for (int it = 0; it < 8; ++it) {
            int i = tid + it * 256;
            int r = i >> 5, c = i & 31;
            __builtin_amdgcn_global_load_async_to_lds_b128(
                (gv4i_ptr)(src + i), (lv4i_ptr)&sC[buf][r * CPAD + c * 4], 0, 0);
        }
        if (tid < CHUNK) sC2[buf][tid] = c2[c0 + tid];
    };

    stage(0, 0);  // prologue: chunk 0 in flight

    for (int c = 0; c < NCHUNK; ++c) {
        const int cur = c & 1;
        if (c + 1 < NCHUNK) {
            stage((c + 1) * CHUNK, 1 - cur);      // prefetch next chunk (other buffer)
            __builtin_amdgcn_s_wait_asynccnt(8);  // chunk c's 8 async ops have landed
        } else {
            __builtin_amdgcn_s_wait_asynccnt(0);
        }
        __syncthreads();                          // all waves' staging visible

        // ---- compute: 4 independent WMMA accumulator chains (one per N-tile) ----
        v8f a0 = {0.f,0.f,0.f,0.f,0.f,0.f,0.f,0.f};
        v8f a1 = a0, a2 = a0, a3 = a0;
        const float* bbase = &sC[cur][laneLo * CPAD + (hi ? 2 : 0)];
#pragma unroll
        for (int k = 0; k < 32; ++k) {
            v2f b0 = *(const v2f*)(bbase + 0 * 16 * CPAD + 4 * k);  // ds_load_b64
            v2f b1 = *(const v2f*)(bbase + 1 * 16 * CPAD + 4 * k);
            v2f b2 = *(const v2f*)(bbase + 2 * 16 * CPAD + 4 * k);
            v2f b3 = *(const v2f*)(bbase + 3 * 16 * CPAD + 4 * k);
            a0 = __builtin_amdgcn_wmma_f32_16x16x4_f32(false, ax[k], false, b0, (short)0, a0, false, false);
            a1 = __builtin_amdgcn_wmma_f32_16x16x4_f32(false, ax[k], false, b1, (short)0, a1, false, false);
            a2 = __builtin_amdgcn_wmma_f32_16x16x4_f32(false, ax[k], false, b2, (short)0, a2, false, false);
            a3 = __builtin_amdgcn_wmma_f32_16x16x4_f32(false, ax[k], false, b3, (short)0, a3, false, false);
        }

        // ---- scoring: score = ||c||^2 - 2 x.c (monotone in distance) ----
        const int c0id = c * CHUNK + laneLo;
#pragma unroll
        for (int t = 0; t < 4; ++t) {
            const v8f   acc = (t == 0) ? a0 : (t == 1) ? a1 : (t == 2) ? a2 : a3;
            const int   cid = c0id + t * 16;
            const float c2v = sC2[cur][t * 16 + laneLo];
#pragma unroll
            for (int j = 0; j < 8; ++j) {
                float score = c2v - 2.0f * acc[j];
                if (score < bestV[j]) { bestV[j] = score; bestI[j] = cid; }
            }
        }
        __syncthreads();  // compute on buf `cur` done -> safe to refill it next iter
    }

    // ---- cross-lane argmin within each 16-lane half (rows 0-7 / 8-15) ----
#pragma unroll
    for (int d = 1; d < 16; d <<= 1) {
#pragma unroll
        for (int j = 0; j < 8; ++j) {
            float ov = __shfl_xor(bestV[j], d, 32);
            int   oi = __shfl_xor(bestI[j], d, 32);
            if (ov < bestV[j] || (ov == bestV[j] && oi < bestI[j])) {
                bestV[j] = ov; bestI[j] = oi;   // first-occurrence tie-break
            }
        }
    }
    if (laneLo == 0) {  // lane 0 -> rows 0..7, lane 16 -> rows 8..15
#pragma unroll
        for (int j = 0; j < 8; ++j) {
            int row = j + (hi ? 8 : 0);
            sFinV[wave * 16 + row] = bestV[j];
            sFinI[wave * 16 + row] = bestI[j];
        }
    }
    __syncthreads();

    if (lane < 16) {
        const int   gRow = rowBase + laneLo;
        const float bv   = sFinV[wave * 16 + laneLo];
        const int   bi   = sFinI[wave * 16 + laneLo];
        float sq = x2 + bv;                 // lane<16 holds x2 of row laneLo
        sq = sq < 0.f ? 0.f : sq;
        outMin[gRow] = sqrtf(sq);
        outClu[gRow] = (float)(bi % KC);
        outCls[gRow] = (float)(bi / KC);
    }
    __syncthreads();

    // ---- gather winning center rows: 128 floats per row = one float4 per lane ----
#pragma unroll 1
    for (int r = 0; r < 16; ++r) {
        int bi   = sFinI[wave * 16 + r];
        int gRow = rowBase + r;
        float4 v = ((const float4*)(centers + (size_t)bi * DIM))[lane];
        ((float4*)(outQ + (size_t)gRow * DIM))[lane] = v;
    }
}

// ---------------------------------------------------------------------------
extern "C" void kernel_launch(void* const* d_in, const int* in_sizes, int n_in,
                              void* d_out, int out_size, void* d_ws, size_t ws_size,
                              hipStream_t stream) {
    const float* x       = (const float*)d_in[0];  // [32768,128] f32
    const float* centers = (const float*)d_in[1];  // [100,64,128] f32
    // d_in[2] = labels (unused by the forward outputs)

    float* c2     = (float*)d_ws;                  // 6400 floats scratch
    float* outQ   = (float*)d_out;                 // [B,128]
    float* outClu = outQ + (size_t)BATCH * DIM;    // [B]
    float* outMin = outClu + BATCH;                // [B]
    float* outCls = outMin + BATCH;                // [B]

    c2_kernel<<<(K_CENTERS + 255) / 256, 256, 0, stream>>>(centers, c2);
    vq_kernel<<<BATCH / BLOCK_ROWS, 256, 0, stream>>>(x, centers, c2,
                                                      outQ, outClu, outMin, outCls);
}